// GcnNet_14705968022298
// MI455X (gfx1250) — compile-verified
//
#include <hip/hip_runtime.h>

typedef __attribute__((ext_vector_type(2))) float v2f;
typedef __attribute__((ext_vector_type(8))) float v8f;

#define NNODES 50000
#define INDIM  512
#define HIDDIM 128
#define OUTDIM 16

// ---------------------------------------------------------------------------
// GEMM1: xw1[50000,128] = feat[50000,512] @ W1[512,128]  (f32 WMMA 16x16x4)
// Block = 256 threads (8 waves). Block computes 16 rows x 128 cols.
// Feature tile staged transposed in LDS: As[k][m], padded stride 17
// (conflict-free A-fragment reads).
// ---------------------------------------------------------------------------
__global__ __launch_bounds__(256) void gemm1_wmma(const float* __restrict__ feat,
                                                  const float* __restrict__ W1,
                                                  float* __restrict__ xw1) {
  __shared__ float As[INDIM][17];  // [k][m] transposed, padded
  const int row0 = blockIdx.x * 16;
  const int tid  = threadIdx.x;

  // Cooperative load: 16 rows x 512 cols = 2048 float4, 8 per thread,
  // coalesced along k within each row.
  const float4* src = (const float4*)(feat + (size_t)row0 * INDIM);
#pragma unroll
  for (int i = 0; i < 8; ++i) {
    const int s  = tid + i * 256;   // 0..2047
    const int r  = s >> 7;          // row within tile (0..15)
    const int kv = s & 127;         // float4 index along k
    const float4 v = src[(size_t)r * (INDIM / 4) + kv];
    const int k = kv * 4;
    As[k + 0][r] = v.x;
    As[k + 1][r] = v.y;
    As[k + 2][r] = v.z;
    As[k + 3][r] = v.w;
  }
  __syncthreads();

  const int wave = tid >> 5;
  const int lane = tid & 31;
  const int hs   = lane >> 4;   // half-wave select (K offset 0 or 2)
  const int lm   = lane & 15;   // M for A-frag, N for B/C/D frags
  const int n0   = wave * 16;   // this wave's 16-column slice of W1

  v8f acc = {};
  for (int k0 = 0; k0 < INDIM; k0 += 4) {
    const int ka = k0 + hs * 2;
    v2f a, b;
    // A 16x4 f32 layout: lanes 0-15 -> K=k0,k0+1 ; lanes 16-31 -> K=k0+2,k0+3
    a.x = As[ka + 0][lm];
    a.y = As[ka + 1][lm];
    // B 4x16 f32 layout: row (K) striped across VGPRs, N across lanes
    const float* w = W1 + (size_t)ka * HIDDIM + n0 + lm;
    b.x = w[0];
    b.y = w[HIDDIM];
    acc = __builtin_amdgcn_wmma_f32_16x16x4_f32(false, a, false, b,
                                                (short)0, acc, false, false);
  }

  // C/D layout: VGPR j -> M = j + 8*hs, N = lm
  float* dst = xw1 + (size_t)row0 * HIDDIM + n0 + lm;
#pragma unroll
  for (int j = 0; j < 8; ++j)
    dst[(size_t)(hs * 8 + j) * HIDDIM] = acc[j];
}

// ---------------------------------------------------------------------------
// SpMM1: h[dst] += val * xw1[src]   (D = 128, one wave per edge,
// lane carries float4 slice; f32 atomics land in L2 — h is 25.6MB, L2-resident)
// ---------------------------------------------------------------------------
__global__ __launch_bounds__(256) void spmm1_atomic(const int* __restrict__ esrc,
                                                    const int* __restrict__ edst,
                                                    const float* __restrict__ evals,
                                                    const float* __restrict__ xw1,
                                                    float* __restrict__ h,
                                                    int nedges) {
  const int lane   = threadIdx.x & 31;
  const int gwave  = (int)((blockIdx.x * blockDim.x + threadIdx.x) >> 5);
  const int nwaves = (int)((gridDim.x * blockDim.x) >> 5);
  for (int e = gwave; e < nedges; e += nwaves) {
    const int   s = esrc[e];
    const int   d = edst[e];
    const float v = evals[e];
    const float4 x = ((const float4*)(xw1 + (size_t)s * HIDDIM))[lane];
    float* out = h + (size_t)d * HIDDIM + lane * 4;
    atomicAdd(out + 0, v * x.x);
    atomicAdd(out + 1, v * x.y);
    atomicAdd(out + 2, v * x.z);
    atomicAdd(out + 3, v * x.w);
  }
}

// ---------------------------------------------------------------------------
// GEMM2: hw2[50000,16] = relu(h)[50000,128] @ W2[128,16]  (f32 WMMA 16x16x4)
// One wave per 16-row tile; relu fused into A-frag load; W2 (8KB) from L2.
// ---------------------------------------------------------------------------
__global__ __launch_bounds__(32) void gemm2_wmma(const float* __restrict__ h,
                                                 const float* __restrict__ W2,
                                                 float* __restrict__ hw2) {
  const int row0 = blockIdx.x * 16;
  const int lane = threadIdx.x & 31;
  const int hs   = lane >> 4;
  const int lm   = lane & 15;

  const float* arow = h + (size_t)(row0 + lm) * HIDDIM;
  v8f acc = {};
  for (int k0 = 0; k0 < HIDDIM; k0 += 4) {
    const int ka = k0 + hs * 2;
    v2f a, b;
    a.x = fmaxf(arow[ka + 0], 0.0f);
    a.y = fmaxf(arow[ka + 1], 0.0f);
    const float* w = W2 + (size_t)ka * OUTDIM + lm;
    b.x = w[0];
    b.y = w[OUTDIM];
    acc = __builtin_amdgcn_wmma_f32_16x16x4_f32(false, a, false, b,
                                                (short)0, acc, false, false);
  }
  float* dst = hw2 + (size_t)row0 * OUTDIM + lm;
#pragma unroll
  for (int j = 0; j < 8; ++j)
    dst[(size_t)(hs * 8 + j) * OUTDIM] = acc[j];
}

// ---------------------------------------------------------------------------
// SpMM2: out[dst] += val * hw2[src]   (D = 16, 16 threads per edge)
// ---------------------------------------------------------------------------
__global__ __launch_bounds__(256) void spmm2_atomic(const int* __restrict__ esrc,
                                                    const int* __restrict__ edst,
                                                    const float* __restrict__ evals,
                                                    const float* __restrict__ hw2,
                                                    float* __restrict__ out,
                                                    int nedges) {
  const int t      = (int)(blockIdx.x * blockDim.x + threadIdx.x);
  const int stride = (int)((gridDim.x * blockDim.x) >> 4);
  const int d      = t & 15;
  for (int e = t >> 4; e < nedges; e += stride) {
    const int   s  = esrc[e];
    const int   dd = edst[e];
    const float v  = evals[e];
    atomicAdd(out + (size_t)dd * OUTDIM + d, v * hw2[(size_t)s * OUTDIM + d]);
  }
}

// ---------------------------------------------------------------------------
// Launch. Inputs: 0=feature 1=edge_src 2=edge_dst 3=edge_vals 4=W1 5=W2
// Workspace: [0, 25.6MB) = xw1 (later reused as hw2), [25.6MB, 51.2MB) = h
// ---------------------------------------------------------------------------
extern "C" void kernel_launch(void* const* d_in, const int* in_sizes, int n_in,
                              void* d_out, int out_size, void* d_ws, size_t ws_size,
                              hipStream_t stream) {
  const float* feat  = (const float*)d_in[0];
  const int*   esrc  = (const int*)d_in[1];
  const int*   edst  = (const int*)d_in[2];
  const float* evals = (const float*)d_in[3];
  const float* W1    = (const float*)d_in[4];
  const float* W2    = (const float*)d_in[5];
  float*       out   = (float*)d_out;
  const int nedges   = in_sizes[1];

  const size_t xw1_bytes = (size_t)NNODES * HIDDIM * sizeof(float);  // 25.6 MB
  float* xw1 = (float*)d_ws;
  float* h   = (float*)((char*)d_ws + xw1_bytes);
  float* hw2 = (float*)d_ws;  // reuse xw1 space (dead after spmm1)

  // zero the accumulators (graph-capturable async memsets)
  hipMemsetAsync(h, 0, xw1_bytes, stream);
  hipMemsetAsync(out, 0, (size_t)NNODES * OUTDIM * sizeof(float), stream);

  // 1) xw1 = feat @ W1        (50000/16 = 3125 exact tiles)
  gemm1_wmma<<<NNODES / 16, 256, 0, stream>>>(feat, W1, xw1);

  // 2) h = segment_sum(val * xw1[src])
  spmm1_atomic<<<16384, 256, 0, stream>>>(esrc, edst, evals, xw1, h, nedges);

  // 3) hw2 = relu(h) @ W2
  gemm2_wmma<<<NNODES / 16, 32, 0, stream>>>(h, W2, hw2);

  // 4) out = segment_sum(val * hw2[src])
  spmm2_atomic<<<16384, 256, 0, stream>>>(esrc, edst, evals, hw2, out, nedges);
}